// E8P12RVQ3B_codebook_2430951490131
// MI455X (gfx1250) — compile-verified
//
#include <hip/hip_runtime.h>
#include <stdint.h>

// CDNA5 / gfx1250 E8P quantizer.
// Core op: argmax_k ( 2 * x . g_k - |g_k|^2 ) over codebooks, done with
// V_WMMA_F32_16X16X4_F32 (two chained WMMAs for K=8), f32 end to end.

typedef float v2f __attribute__((ext_vector_type(2)));
typedef float v8f __attribute__((ext_vector_type(8)));

#define WAVES 8          // 256 threads = 8 wave32
#define KTMAX 87         // max grid tiles of 16 rows (1366 real rows -> 86 tiles)

// wave-internal LDS store->load handoff (CDNA5 split counters)
#define LDS_SYNC() asm volatile("s_wait_dscnt 0x0" ::: "memory")

__device__ __forceinline__ void stage_xpart(float* ar, const float* x, float shift) {
    float xq[8]; int nc = 0;
#pragma unroll
    for (int i = 0; i < 8; ++i) { xq[i] = x[i] + shift; nc += (xq[i] < 0.f) ? 1 : 0; }
    float xp[8];
#pragma unroll
    for (int i = 0; i < 8; ++i) xp[i] = fabsf(xq[i]);
    if (nc & 1) xp[7] = -xp[7];
#pragma unroll
    for (int i = 0; i < 8; ++i) ar[i] = 2.f * xp[i];   // pre-scale by 2 for score trick
}

// One full argmax sweep: scores S = A(16x8) * G^T(8x16) tiles via 2x WMMA f32 16x16x4,
// C preloaded with -|g|^2 (negated norms staged in LDS). Per-lane running argmax +
// shfl_xor reduce over the 16-lane N-groups (wave32 halves stay separate).
__device__ __forceinline__ void argmax_sweep(const float* __restrict__ sG,
                                             const float* __restrict__ sNneg,
                                             int ntiles, const float* aBase,
                                             int lane, int* redOut) {
    const int m  = lane & 15;
    const int k0 = (lane >> 4) << 1;   // lanes 0-15: K=0,1 ; lanes 16-31: K=2,3
    const float* ar = aBase + m * 8;
    v2f a0, a1;
    a0.x = ar[k0];     a0.y = ar[k0 + 1];
    a1.x = ar[4 + k0]; a1.y = ar[5 + k0];

    float best[8]; int bidx[8];
#pragma unroll
    for (int j = 0; j < 8; ++j) { best[j] = -3.0e38f; bidx[j] = 0; }

#pragma unroll 2
    for (int t = 0; t < ntiles; ++t) {
        const int ng = t * 16 + m;
        const float* g = sG + (size_t)ng * 8;
        v2f b0, b1;
        b0.x = g[k0];     b0.y = g[k0 + 1];
        b1.x = g[4 + k0]; b1.y = g[5 + k0];
        const float nv = sNneg[ng];      // already -|g|^2
        v8f c;
#pragma unroll
        for (int j = 0; j < 8; ++j) c[j] = nv;       // C = -|g|^2 broadcast over M
        // D = A*B + C  (K=8 as two K=4 steps) => 2 x.g - |g|^2
        c = __builtin_amdgcn_wmma_f32_16x16x4_f32(false, a0, false, b0, (short)0, c, false, false);
        c = __builtin_amdgcn_wmma_f32_16x16x4_f32(false, a1, false, b1, (short)0, c, false, false);
#pragma unroll
        for (int j = 0; j < 8; ++j)
            if (c[j] > best[j]) { best[j] = c[j]; bidx[j] = ng; }  // idx monotone -> first max kept
    }
    // reduce across the 16 lanes holding the same M rows (xor<16 keeps halves separate)
#pragma unroll
    for (int off = 1; off < 16; off <<= 1) {
#pragma unroll
        for (int j = 0; j < 8; ++j) {
            float so = __shfl_xor(best[j], off, 32);
            int   io = __shfl_xor(bidx[j], off, 32);
            if (so > best[j] || (so == best[j] && io < bidx[j])) { best[j] = so; bidx[j] = io; }
        }
    }
    if ((lane & 15) == 0) {
        const int half = (lane >> 4) * 8;    // lane 0 -> rows 0..7, lane 16 -> rows 8..15
#pragma unroll
        for (int j = 0; j < 8; ++j) redOut[half + j] = bidx[j];
    }
}

__device__ __forceinline__ void eval_pass(const float* x, float shift, int parity, int idx,
                                          const float* __restrict__ sGrid,
                                          const int* __restrict__ pam,
                                          const unsigned char* __restrict__ gao,
                                          float* vals, float* err2o, int* packed) {
    float xq[8]; bool neg[8]; int nc = 0;
#pragma unroll
    for (int i = 0; i < 8; ++i) { xq[i] = x[i] + shift; neg[i] = xq[i] < 0.f; nc += neg[i] ? 1 : 0; }
    float mask[8];
#pragma unroll
    for (int i = 0; i < 8; ++i) mask[i] = neg[i] ? -1.f : 1.f;
    if (nc & 1) mask[7] = -mask[7];
    const float* g = &sGrid[(size_t)idx * 8];
    float e2 = 0.f; int sm[8];
#pragma unroll
    for (int i = 0; i < 8; ++i) {
        float v = g[i] * mask[i];
        vals[i] = v;
        float d = xq[i] - v;
        e2 += d * d;
        sm[i] = ((g[i] < 0.f) != (mask[i] < 0.f)) ? 1 : 0;
    }
    const int ai = pam[idx];
    const int PERM[8] = {0, 2, 4, 6, 1, 3, 5, 7};
    int bits = 0;
#pragma unroll
    for (int j = 0; j < 8; ++j) {
        int b = sm[PERM[j]];
        if (j == 7) b ^= (gao[ai] & 1);
        if (j == 0) b ^= parity;
        bits |= b << j;
    }
    *packed = (ai << 8) | bits;
    *err2o = e2;
}

__global__ void __launch_bounds__(256)
e8p_quant_kernel(const float* __restrict__ X,
                 const float* __restrict__ gp,  const float* __restrict__ gpn,
                 const float* __restrict__ e8,  const float* __restrict__ e8n,
                 const int* __restrict__ pam,   const unsigned char* __restrict__ gao,
                 float* __restrict__ outV, unsigned long long* __restrict__ outI,
                 int N, int Kp, int KT) {
    __shared__ float sGrid[KTMAX * 16 * 8];
    __shared__ float sNormNeg[KTMAX * 16];
    __shared__ float sE8[256 * 8];
    __shared__ float sE8nNeg[256];
    __shared__ float sA[WAVES][16 * 8];
    __shared__ int   sIdxP[WAVES][16];
    __shared__ int   sIdxM[WAVES][16];
    __shared__ int   sIdxR[WAVES][16];

    const int tid = threadIdx.x;
    const int gridElems = KT * 16 * 8;
    for (int i = tid; i < gridElems; i += 256) {
        int row = i >> 3;
        sGrid[i] = (row < Kp) ? gp[i] : 0.f;
    }
    for (int i = tid; i < KT * 16; i += 256)
        sNormNeg[i] = (i < Kp) ? -gpn[i] : -3.0e38f;     // padded rows never win
    for (int i = tid; i < 256 * 8; i += 256) sE8[i] = e8[i];
    for (int i = tid; i < 256; i += 256) sE8nNeg[i] = -e8n[i];
    __syncthreads();

    const int wave = tid >> 5, lane = tid & 31;
    float* aRowBase = &sA[wave][0];
    const int tilesTotal = (N + 15) >> 4;
    const int tileStride = (int)gridDim.x * WAVES;

    for (int tile = blockIdx.x * WAVES + wave; tile < tilesTotal; tile += tileStride) {
        const int base = tile * 16;
        const int row  = base + (lane & 15);
        float x[8];
#pragma unroll
        for (int i = 0; i < 8; ++i) x[i] = 0.f;

        if (lane < 16) {
            if (row < N) {
#pragma unroll
                for (int i = 0; i < 8; ++i) x[i] = X[(size_t)row * 8 + i];
            }
            // warm L2/L0 for the next grid-stride tile (global_prefetch_b8)
            const int nrow = row + tileStride * 16;
            if (nrow < N) __builtin_prefetch(&X[(size_t)nrow * 8], 0, 0);
            stage_xpart(aRowBase + lane * 8, x, +0.25f);     // plus pass A fragment
        }
        LDS_SYNC();
        argmax_sweep(sGrid, sNormNeg, KT, aRowBase, lane, sIdxP[wave]);
        LDS_SYNC();
        if (lane < 16) stage_xpart(aRowBase + lane * 8, x, -0.25f);  // minus pass
        LDS_SYNC();
        argmax_sweep(sGrid, sNormNeg, KT, aRowBase, lane, sIdxM[wave]);
        LDS_SYNC();

        float initVals[8]; int initIdx = 0;
        if (lane < 16) {
            const int ip = sIdxP[wave][lane & 15];
            const int im = sIdxM[wave][lane & 15];
            float valsP[8], valsM[8], eP, eM; int pkP, pkM;
            eval_pass(x, +0.25f, 1, ip, sGrid, pam, gao, valsP, &eP, &pkP);
            eval_pass(x, -0.25f, 0, im, sGrid, pam, gao, valsM, &eM, &pkM);
            const bool useP = eP < eM;
#pragma unroll
            for (int i = 0; i < 8; ++i)
                initVals[i] = useP ? (valsP[i] - 0.25f) : (valsM[i] + 0.25f);
            initIdx = useP ? pkP : pkM;
            float* ar = aRowBase + lane * 8;
#pragma unroll
            for (int i = 0; i < 8; ++i)
                ar[i] = 2.f * ((x[i] - initVals[i]) * 2.04f);   // residual, pre-scaled
        }
        LDS_SYNC();
        argmax_sweep(sE8, sE8nNeg, 16, aRowBase, lane, sIdxR[wave]);  // 256-entry residual book
        LDS_SYNC();
        if (lane < 16 && row < N) {
            const int ridx = sIdxR[wave][lane & 15];
            const float* rv = &sE8[(size_t)ridx * 8];
#pragma unroll
            for (int i = 0; i < 8; ++i)
                outV[(size_t)row * 8 + i] = initVals[i] + rv[i] * (1.0f / 2.04f);
            outI[row] = ((unsigned long long)(unsigned)initIdx << 16) | (unsigned)ridx;
        }
    }
}

extern "C" void kernel_launch(void* const* d_in, const int* in_sizes, int n_in,
                              void* d_out, int out_size, void* d_ws, size_t ws_size,
                              hipStream_t stream) {
    (void)n_in; (void)out_size; (void)d_ws; (void)ws_size;
    const float* X   = (const float*)d_in[0];
    const float* gp  = (const float*)d_in[1];
    const float* gpn = (const float*)d_in[2];
    const float* e8  = (const float*)d_in[3];
    const float* e8n = (const float*)d_in[4];
    const int*   pam = (const int*)d_in[5];
    const unsigned char* gao = (const unsigned char*)d_in[6];

    const int N  = in_sizes[0] / 8;
    const int Kp = in_sizes[1] / 8;
    int KT = (Kp + 15) / 16;
    if (KT > KTMAX) KT = KTMAX;

    float* outV = (float*)d_out;
    unsigned long long* outI = (unsigned long long*)(outV + (size_t)N * 8);

    const int tiles = (N + 15) / 16;
    int blocks = (tiles + WAVES - 1) / WAVES;
    if (blocks > 2048) blocks = 2048;
    if (blocks < 1) blocks = 1;

    e8p_quant_kernel<<<blocks, 256, 0, stream>>>(X, gp, gpn, e8, e8n, pam, gao,
                                                 outV, outI, N, Kp, KT);
}